// FilterProposals_22703197127181
// MI455X (gfx1250) — compile-verified
//
#include <hip/hip_runtime.h>
#include <hip/hip_bf16.h>

typedef __attribute__((ext_vector_type(2))) float v2f;
typedef __attribute__((ext_vector_type(8))) float v8f;
typedef __attribute__((ext_vector_type(4))) unsigned tdm_u4;
typedef __attribute__((ext_vector_type(4))) int      tdm_i4;
typedef __attribute__((ext_vector_type(8))) int      tdm_i8;

#define N_IMG     8
#define A_TOTAL   65472
#define K_TOTAL   6960
#define POST_NMS  2000
#define NSORT     8192
#define NW        218          // ceil(K_TOTAL / 32)
#define NMS_TH    0.7f
#define MIN_SIZE  1e-3f
#define IMG_SZ    512.0f
#define LVL_OFF   600.0f       // > max clipped coord; separates levels exactly like ref offset

__constant__ int c_lvlN[5]    = {49152, 12288, 3072, 768, 192};
__constant__ int c_lvlK[5]    = {2000, 2000, 2000, 768, 192};
__constant__ int c_lvlAOff[5] = {0, 49152, 61440, 64512, 65280};
__constant__ int c_lvlBase[5] = {0, 2000, 4000, 6000, 6768};

__device__ __forceinline__ unsigned fkey(float f) {
    unsigned u = __float_as_uint(f);
    return (u & 0x80000000u) ? ~u : (u | 0x80000000u);   // ascending-order key
}

// ---- Tensor Data Mover: 1-D global->LDS DMA (D# per CDNA5 ISA ch.8) ----------------
// nelem = number of 4-byte elements (< 65536 so it fits tile_dim0).
__device__ __forceinline__ void tdm_copy_to_lds(unsigned lds_off, const void* gptr,
                                                unsigned nelem)
{
    unsigned long long ga = (unsigned long long)(size_t)gptr;
    tdm_u4 g0;
    g0[0] = 1u;                                             // count=1 (valid), user mode
    g0[1] = lds_off;                                        // lds_addr (bytes)
    g0[2] = (unsigned)(ga & 0xFFFFFFFFull);                 // global_addr[31:0]
    g0[3] = (unsigned)((ga >> 32) & 0x01FFFFFFull) | (2u << 30); // addr[56:32] | type=2
    tdm_i8 g1;
    g1[0] = (int)(2u << 16);                                // wg_mask=0, data_size=2 (4B)
    g1[1] = (int)((nelem & 0xFFFFu) << 16);                 // tensor_dim0[15:0]  @ bits 63:48
    g1[2] = (int)((nelem >> 16) & 0xFFFFu);                 // tensor_dim0[31:16]; tensor_dim1=0
    g1[3] = (int)((nelem & 0xFFFFu) << 16);                 // tile_dim0 @ bits 127:112
    g1[4] = 0;                                              // tile_dim1=0, tile_dim2=0
    g1[5] = (int)nelem;                                     // tensor_dim0_stride[31:0]
    g1[6] = 0;                                              // stride[47:32]=0, dim1_stride=0
    g1[7] = 0;
    tdm_i4 gz = {0, 0, 0, 0};
#if __clang_major__ >= 23
    tdm_i8 g4 = {0, 0, 0, 0, 0, 0, 0, 0};
    __builtin_amdgcn_tensor_load_to_lds(g0, g1, gz, gz, g4, 0);
#else
    __builtin_amdgcn_tensor_load_to_lds(g0, g1, gz, gz, 0);
#endif
}

__device__ __forceinline__ void emit_candidate(
    const float* __restrict__ proposals, const float* __restrict__ objRow,
    int img, int aoff, int i, int slot,
    float* __restrict__ selBox, float* __restrict__ selScore, int* __restrict__ selValid)
{
    float o  = objRow[i];
    float sc = 1.0f / (1.0f + __expf(-o));
    const float* p = proposals + ((size_t)img * A_TOTAL + (size_t)aoff + i) * 4;
    float x1 = fminf(fmaxf(p[0], 0.0f), IMG_SZ);
    float y1 = fminf(fmaxf(p[1], 0.0f), IMG_SZ);
    float x2 = fminf(fmaxf(p[2], 0.0f), IMG_SZ);
    float y2 = fminf(fmaxf(p[3], 0.0f), IMG_SZ);
    int valid = ((x2 - x1) >= MIN_SIZE) && ((y2 - y1) >= MIN_SIZE) && (sc >= 0.0f);
    size_t sb = ((size_t)img * K_TOTAL + slot);
    selBox[sb * 4 + 0] = x1; selBox[sb * 4 + 1] = y1;
    selBox[sb * 4 + 2] = x2; selBox[sb * 4 + 3] = y2;
    selScore[sb] = sc;
    selValid[sb] = valid;
}

// ---------------- Stage A: exact per-(image,level) top-k via 3-round radix select ----
__global__ __launch_bounds__(1024) void topk_gather_kernel(
    const float* __restrict__ proposals, const float* __restrict__ objectness,
    float* __restrict__ selBox, float* __restrict__ selScore, int* __restrict__ selValid)
{
    __shared__ int hist[4096];
    __shared__ unsigned sPrefix;
    __shared__ int sNeed, sSlot, sTie;

    const int img  = blockIdx.x / 5;
    const int lvl  = blockIdx.x % 5;
    const int n    = c_lvlN[lvl];
    const int k    = c_lvlK[lvl];
    const int aoff = c_lvlAOff[lvl];
    const int base = c_lvlBase[lvl];
    const int tid  = threadIdx.x;
    const int nthr = blockDim.x;
    const float* objRow = objectness + (size_t)img * A_TOTAL + aoff;

    if (k == n) {   // small levels: take everything
        for (int i = tid; i < n; i += nthr)
            emit_candidate(proposals, objRow, img, aoff, i, base + i, selBox, selScore, selValid);
        return;
    }

    // pre-zero our slot range (insurance against poisoned workspace)
    for (int s = tid; s < k; s += nthr) {
        size_t sb = (size_t)img * K_TOTAL + base + s;
        selBox[sb * 4 + 0] = 0.f; selBox[sb * 4 + 1] = 0.f;
        selBox[sb * 4 + 2] = 0.f; selBox[sb * 4 + 3] = 0.f;
        selScore[sb] = 0.f; selValid[sb] = 0;
    }

    unsigned prefix = 0;
    int need = k;
    const int shifts[3] = {20, 8, 0};
    const int nbuck[3]  = {4096, 4096, 256};
    for (int rnd = 0; rnd < 3; ++rnd) {
        const int sh = shifts[rnd], nb = nbuck[rnd];
        for (int b = tid; b < 4096; b += nthr) hist[b] = 0;
        __syncthreads();
        for (int i = tid; i < n; i += nthr) {
            __builtin_prefetch(objRow + i + 8192, 0, 0);     // global_prefetch_b8
            unsigned kk = fkey(objRow[i]);
            bool match = (rnd == 0) ||
                         (rnd == 1 && (kk >> 20) == (prefix >> 20)) ||
                         (rnd == 2 && (kk >> 8)  == (prefix >> 8));
            if (match) atomicAdd(&hist[(kk >> sh) & (nb - 1)], 1);
        }
        __syncthreads();
        if (tid == 0) {
            int acc = 0, nd = need; unsigned pb = 0;
            for (int b = nb - 1; b >= 0; --b) {
                int h = hist[b];
                if (acc + h >= nd) { pb = (unsigned)b; nd -= acc; break; }
                acc += h;
            }
            sPrefix = prefix | (pb << sh);
            sNeed = nd;
        }
        __syncthreads();
        prefix = sPrefix; need = sNeed;
        __syncthreads();
    }
    const unsigned T = prefix;
    const int needT = need;
    if (tid == 0) { sSlot = 0; sTie = 0; }
    __syncthreads();
    for (int i = tid; i < n; i += nthr) {
        unsigned kk = fkey(objRow[i]);
        bool take = kk > T;
        if (!take && kk == T) take = (atomicAdd(&sTie, 1) < needT);
        if (take) {
            int slot = base + atomicAdd(&sSlot, 1);
            if (slot < base + k)
                emit_candidate(proposals, objRow, img, aoff, i, slot, selBox, selScore, selValid);
        }
    }
}

// ---------------- Stage B: per-image bitonic sort of 8192 (score desc, slot asc) -----
__global__ __launch_bounds__(1024) void sort_kernel(
    const float* __restrict__ selBox, const float* __restrict__ selScore,
    const int* __restrict__ selValid,
    float* __restrict__ sBox, float* __restrict__ sScore,
    int* __restrict__ sValid, int* __restrict__ sLvl)
{
    extern __shared__ unsigned char smB[];
    float* skey = (float*)smB;
    int*   sidx = (int*)(smB + NSORT * sizeof(float));
    const int img = blockIdx.x, tid = threadIdx.x, nthr = blockDim.x;
    const size_t ib = (size_t)img * K_TOTAL;
    const float NEG = -__builtin_inff();

    for (int s = tid; s < NSORT; s += nthr) {
        if (s < K_TOTAL) { skey[s] = selValid[ib + s] ? selScore[ib + s] : NEG; sidx[s] = s; }
        else             { skey[s] = NEG; sidx[s] = s; }   // pads sort after real entries
    }
    for (unsigned size = 2; size <= NSORT; size <<= 1) {
        for (unsigned stride = size >> 1; stride > 0; stride >>= 1) {
            __syncthreads();
            for (unsigned t = tid; t < NSORT / 2; t += nthr) {
                unsigned a = 2u * t - (t & (stride - 1));
                unsigned b = a + stride;
                bool desc = ((a & size) == 0);
                float ka = skey[a], kb = skey[b];
                int ia = sidx[a], ibn = sidx[b];
                bool aFirst = (ka > kb) || (ka == kb && ia < ibn);
                if (aFirst != desc) { skey[a] = kb; skey[b] = ka; sidx[a] = ibn; sidx[b] = ia; }
            }
        }
    }
    __syncthreads();
    for (int s = tid; s < K_TOTAL; s += nthr) {
        int slot = sidx[s]; if (slot >= K_TOTAL) slot = K_TOTAL - 1;
        size_t src = ib + slot, dst = ib + s;
        sBox[dst * 4 + 0] = selBox[src * 4 + 0];
        sBox[dst * 4 + 1] = selBox[src * 4 + 1];
        sBox[dst * 4 + 2] = selBox[src * 4 + 2];
        sBox[dst * 4 + 3] = selBox[src * 4 + 3];
        sScore[dst] = selScore[src];
        sValid[dst] = selValid[src];
        sLvl[dst]   = slot < 2000 ? 0 : slot < 4000 ? 1 : slot < 6000 ? 2 : slot < 6768 ? 3 : 4;
    }
}

// ---------------- Stage C: TDM-staged, WMMA-assisted chunked bitmask NMS -------------
__global__ __launch_bounds__(1024) void nms_out_kernel(
    const float* __restrict__ sBox, const float* __restrict__ sScore,
    const int* __restrict__ sValid, const int* __restrict__ sLvl,
    float* __restrict__ outBoxes, float* __restrict__ outScores)
{
    extern __shared__ unsigned char smC[];
    float4*   lbox  = (float4*)smC;                                   // K * 16 B
    float*    larea = (float*)(smC + (size_t)K_TOTAL * 16);           // K * 4 B
    unsigned* keepW = (unsigned*)(smC + (size_t)K_TOTAL * 20);        // NW words
    unsigned* maskW = keepW + NW;                                     // 16*NW words
    unsigned* wpref = maskW + 16 * NW;                                // NW words
    __shared__ int sTotal;

    const int img = blockIdx.x, tid = threadIdx.x, nthr = blockDim.x;
    const int lane = tid & 31, wave = tid >> 5;
    const size_t ib = (size_t)img * K_TOTAL;

    // TDM DMA: raw sorted boxes (K * 4 floats, contiguous) global -> LDS, wave 0 issues
    if (tid == 0)
        tdm_copy_to_lds((unsigned)(size_t)(void*)lbox, sBox + ib * 4, K_TOTAL * 4);

    // overlap with the DMA: build initial keep bitset from validity flags
    for (int w = tid; w < NW; w += nthr) {
        unsigned word = 0;
        for (int b = 0; b < 32; ++b) {
            int s = w * 32 + b;
            if (s < K_TOTAL && sValid[ib + s]) word |= 1u << b;
        }
        keepW[w] = word;
    }
    if (tid == 0) __builtin_amdgcn_s_wait_tensorcnt(0);
    __syncthreads();

    // in-place transform: area + per-level coordinate offset (level batching)
    for (int s = tid; s < K_TOTAL; s += nthr) {
        float4 b = lbox[s];
        float off = LVL_OFF * (float)sLvl[ib + s];
        larea[s] = (b.z - b.x) * (b.w - b.y);
        lbox[s]  = make_float4(b.x + off, b.y + off, b.z + off, b.w + off);
    }

    for (int c0 = 0; c0 < K_TOTAL; c0 += 16) {       // 435 chunks of 16 rows
        __syncthreads();
        unsigned chunkBits = (keepW[c0 >> 5] >> (c0 & 31)) & 0xFFFFu;
        if (chunkBits == 0) continue;                // dead chunk suppresses nothing
        for (int m = tid; m < 16 * NW; m += nthr) maskW[m] = 0u;
        __syncthreads();

        // A-fragment for rank-2 matmul: D[i][j] = area[c0+i]*1 + 1*area[j]
        v2f afrag;
        if (lane < 16) { afrag[0] = larea[c0 + lane]; afrag[1] = 1.0f; }
        else           { afrag[0] = 0.0f;             afrag[1] = 0.0f; }
        const int Moff = (lane >> 4) << 3;   // C layout: lanes 16-31 hold rows M+8
        const int nn   = lane & 15;

        // chunk-row boxes are invariant across this wave's column tiles: keep in VGPRs
        float4 biArr[8];
        #pragma unroll
        for (int r = 0; r < 8; ++r) biArr[r] = lbox[c0 + r + Moff];

        for (int jt = wave; jt < NW; jt += 32) {     // 16-row x 32-col tiles, one per wave
            const int j0 = jt * 32;
            if (j0 + 31 <= c0) continue;             // upper triangle only
            const int jA = j0 + nn, jB = j0 + 16 + nn;
            v2f bfragA, bfragB;
            if (lane < 16) {
                bfragA[0] = 1.0f; bfragA[1] = (jA < K_TOTAL) ? larea[jA] : 0.0f;
                bfragB[0] = 1.0f; bfragB[1] = (jB < K_TOTAL) ? larea[jB] : 0.0f;
            } else {
                bfragA[0] = bfragA[1] = bfragB[0] = bfragB[1] = 0.0f;
            }
            v8f zero = {0.f, 0.f, 0.f, 0.f, 0.f, 0.f, 0.f, 0.f};
            v8f sumA = __builtin_amdgcn_wmma_f32_16x16x4_f32(
                false, afrag, false, bfragA, (short)0, zero, false, false);
            v8f sumB = __builtin_amdgcn_wmma_f32_16x16x4_f32(
                false, afrag, false, bfragB, (short)0, zero, false, false);

            const float4 bjA = lbox[jA < K_TOTAL ? jA : 0];
            const float4 bjB = lbox[jB < K_TOTAL ? jB : 0];
            #pragma unroll
            for (int r = 0; r < 8; ++r) {
                const int i = c0 + r + Moff;
                const float4 bi = biArr[r];
                float iwA = fminf(bi.z, bjA.z) - fmaxf(bi.x, bjA.x);
                float ihA = fminf(bi.w, bjA.w) - fmaxf(bi.y, bjA.y);
                float inA = fmaxf(iwA, 0.f) * fmaxf(ihA, 0.f);
                // inter > th*(union): division-free, NaN-for-empty matches reference
                bool pA = (jA < K_TOTAL) && (jA > i) && (inA > NMS_TH * (sumA[r] - inA));
                float iwB = fminf(bi.z, bjB.z) - fmaxf(bi.x, bjB.x);
                float ihB = fminf(bi.w, bjB.w) - fmaxf(bi.y, bjB.y);
                float inB = fmaxf(iwB, 0.f) * fmaxf(ihB, 0.f);
                bool pB = (jB < K_TOTAL) && (jB > i) && (inB > NMS_TH * (sumB[r] - inB));
                unsigned bA = __builtin_amdgcn_ballot_w32(pA);
                unsigned bB = __builtin_amdgcn_ballot_w32(pB);
                if (lane == 0) {   // each (row, word) owned by exactly one wave
                    maskW[r * NW + jt]       = (bA & 0xFFFFu) | (bB << 16);
                    maskW[(r + 8) * NW + jt] = (bA >> 16) | (bB & 0xFFFF0000u);
                }
            }
        }
        __syncthreads();
        if (wave == 0) {   // sequential greedy application, 16 rows, one wave
            volatile unsigned* vk = keepW;
            for (int ii = 0; ii < 16; ++ii) {
                const int i = c0 + ii;
                unsigned kw = vk[i >> 5];
                if (kw & (1u << (i & 31))) {
                    for (int w = lane; w < NW; w += 32) {
                        unsigned m = maskW[ii * NW + w];
                        if (m) vk[w] &= ~m;
                    }
                }
            }
        }
    }
    __syncthreads();
    if (tid == 0) {
        int acc = 0;
        for (int w = 0; w < NW; ++w) { wpref[w] = (unsigned)acc; acc += __popc(keepW[w]); }
        sTotal = acc;
    }
    __syncthreads();
    const int kept = sTotal < POST_NMS ? sTotal : POST_NMS;
    float* ob = outBoxes  + (size_t)img * POST_NMS * 4;
    float* os = outScores + (size_t)img * POST_NMS;
    for (int r = kept + tid; r < POST_NMS; r += nthr) {
        ob[r * 4 + 0] = 0.f; ob[r * 4 + 1] = 0.f;
        ob[r * 4 + 2] = 0.f; ob[r * 4 + 3] = 0.f;
        os[r] = 0.f;
    }
    for (int s = tid; s < K_TOTAL; s += nthr) {
        unsigned word = keepW[s >> 5];
        unsigned bit  = 1u << (s & 31);
        if (word & bit) {
            int rank = (int)wpref[s >> 5] + __popc(word & (bit - 1));
            if (rank < POST_NMS) {
                ob[rank * 4 + 0] = sBox[(ib + s) * 4 + 0];
                ob[rank * 4 + 1] = sBox[(ib + s) * 4 + 1];
                ob[rank * 4 + 2] = sBox[(ib + s) * 4 + 2];
                ob[rank * 4 + 3] = sBox[(ib + s) * 4 + 3];
                os[rank] = sScore[ib + s];
            }
        }
    }
}

extern "C" void kernel_launch(void* const* d_in, const int* in_sizes, int n_in,
                              void* d_out, int out_size, void* d_ws, size_t ws_size,
                              hipStream_t stream)
{
    const float* proposals  = (const float*)d_in[0];   // [8, 65472, 4]
    const float* objectness = (const float*)d_in[1];   // [8, 65472]
    (void)in_sizes; (void)n_in; (void)out_size; (void)ws_size;

    float* ws = (float*)d_ws;
    size_t o = 0;
    float* selBox   = ws + o; o += (size_t)N_IMG * K_TOTAL * 4;
    float* selScore = ws + o; o += (size_t)N_IMG * K_TOTAL;
    int*   selValid = (int*)(ws + o); o += (size_t)N_IMG * K_TOTAL;
    float* sBox     = ws + o; o += (size_t)N_IMG * K_TOTAL * 4;
    float* sScore   = ws + o; o += (size_t)N_IMG * K_TOTAL;
    int*   sValid   = (int*)(ws + o); o += (size_t)N_IMG * K_TOTAL;
    int*   sLvl     = (int*)(ws + o); o += (size_t)N_IMG * K_TOTAL;

    float* outBoxes  = (float*)d_out;                          // [8, 2000, 4]
    float* outScores = outBoxes + (size_t)N_IMG * POST_NMS * 4;// [8, 2000]

    topk_gather_kernel<<<N_IMG * 5, 1024, 0, stream>>>(
        proposals, objectness, selBox, selScore, selValid);

    const size_t smB = (size_t)NSORT * 8;                      // 64 KB
    sort_kernel<<<N_IMG, 1024, smB, stream>>>(
        selBox, selScore, selValid, sBox, sScore, sValid, sLvl);

    const size_t smC = (size_t)K_TOTAL * 20 + (size_t)(NW + 16 * NW + NW) * 4 + 16; // ~151 KB
    nms_out_kernel<<<N_IMG, 1024, smC, stream>>>(
        sBox, sScore, sValid, sLvl, outBoxes, outScores);
}